// MVN_DDI_15375982920241
// MI455X (gfx1250) — compile-verified
//
#include <hip/hip_runtime.h>
#include <math.h>
#include <stdint.h>

typedef __attribute__((ext_vector_type(2))) float v2f;
typedef __attribute__((ext_vector_type(8))) float v8f;
typedef unsigned int u32x4 __attribute__((ext_vector_type(4)));
typedef int i32x4 __attribute__((ext_vector_type(4)));
typedef int i32x8 __attribute__((ext_vector_type(8)));

#define N_NODES  50000
#define N_EDGES  1000000
#define IN_DIM   128
#define HEADS    8
#define HEAD_DIM 16
#define OUT_DIM  128           // HEADS*HEAD_DIM
#define WAVES_PER_BLOCK 4

// ---------------------------------------------------------------- zero fill
__global__ void zero_kernel(float* __restrict__ p, int n) {
  int i = blockIdx.x * blockDim.x + threadIdx.x;
  if (i < n) p[i] = 0.0f;
}

// ------------------------------------------------------- TDM weight staging
// Build a Tensor-DMA descriptor (ISA cdna5 §8.3/8.4) for a 2-D 128x128 fp32
// tensor whose tile covers the whole tensor, and DMA it global -> LDS.
// Issued once per block by wave 0; TENSORcnt waited by the issuing wave,
// then a workgroup barrier publishes the LDS contents.
__device__ __forceinline__ void tdm_load_weight_to_lds(const float* gptr,
                                                       float* lds) {
  unsigned long long ga = (unsigned long long)(uintptr_t)gptr;
  // generic LDS address: low 32 bits == LDS byte offset (ISA §10.2)
  unsigned lds_off = (unsigned)(uintptr_t)lds;

  u32x4 g0;
  g0[0] = 1u;                                   // count=1, user descriptor
  g0[1] = lds_off;                              // lds_addr (bytes)
  g0[2] = (unsigned)(ga & 0xffffffffu);         // global_addr[31:0]
  g0[3] = (unsigned)((ga >> 32) & 0x01ffffffu)  // global_addr[56:32]
          | (2u << 30);                         // type=2 ("image")

  i32x8 g1;
  g1[0] = (int)(2u << 16);       // workgroup_mask=0, data_size=2 (4 bytes)
  g1[1] = (int)(128u << 16);     // tensor_dim0[15:0]=128 (elements)
  g1[2] = (int)(128u << 16);     // tensor_dim0 hi=0, tensor_dim1[15:0]=128
  g1[3] = (int)(128u << 16);     // tensor_dim1 hi=0, tile_dim0=128
  g1[4] = 128;                   // tile_dim1=128, tile_dim2=0
  g1[5] = 128;                   // tensor_dim0_stride[31:0]=128 elements
  g1[6] = 0;                     // stride0 hi=0, tensor_dim1_stride lo=0
  g1[7] = 0;                     // tensor_dim1_stride hi=0

  i32x4 zz4 = {0, 0, 0, 0};      // groups 2/3 unused for 2-D tensors
#if __clang_major__ >= 23
  i32x8 zz8 = {0, 0, 0, 0, 0, 0, 0, 0};
  __builtin_amdgcn_tensor_load_to_lds(g0, g1, zz4, zz4, zz8, 0);
#else
  __builtin_amdgcn_tensor_load_to_lds(g0, g1, zz4, zz4, 0);
#endif
  __builtin_amdgcn_s_wait_tensorcnt(0);
}

// ---------------------------------------------------------------- QKV GEMMs
// One wave computes a 16-node tile of Q, K, V.  Weight matrices are staged
// one at a time in 64KB LDS via the TDM (row-major [k][n] -> lanes read
// consecutive n, conflict-free).  fp32 WMMA 16x16x4, K=128 => 32 k-steps.
__global__ __launch_bounds__(128) void qkv_kernel(
    const float* __restrict__ h,
    const float* __restrict__ Wq, const float* __restrict__ Wk,
    const float* __restrict__ Wv,
    float* __restrict__ Q, float* __restrict__ K, float* __restrict__ V)
{
  __shared__ float Wl[IN_DIM * OUT_DIM];   // 64 KB
  const int tid  = threadIdx.x;
  const int lane = tid & 31;
  const int wave = tid >> 5;
  const int m    = lane & 15;              // A row / C col within tile
  const int koff = (lane >> 4) << 1;       // K sub-offset 0 or 2
  const int rrow = (lane >> 4) << 3;       // C row base 0 or 8

  const int tile   = blockIdx.x * WAVES_PER_BLOCK + wave;
  const bool active = (tile * 16) < N_NODES;
  const long arow  = active ? (long)(tile * 16 + m) * IN_DIM : 0;

  const float* Ws[3]  = {Wq, Wk, Wv};
  float*       Os[3]  = {Q,  K,  V};

  #pragma unroll 1
  for (int w = 0; w < 3; ++w) {
    __syncthreads();                        // previous round's compute done
    if (wave == 0) tdm_load_weight_to_lds(Ws[w], Wl);
    __syncthreads();
    if (active) {
      v8f acc[8] = {};
      #pragma unroll 4
      for (int k = 0; k < IN_DIM / 4; ++k) {
        const float2 a2 = *(const float2*)(h + arow + k * 4 + koff);
        v2f a; a[0] = a2.x; a[1] = a2.y;
        const int kb = (k * 4 + koff) * OUT_DIM;
        #pragma unroll
        for (int t = 0; t < 8; ++t) {
          const int col = t * 16 + m;
          v2f b; b[0] = Wl[kb + col]; b[1] = Wl[kb + OUT_DIM + col];
          acc[t] = __builtin_amdgcn_wmma_f32_16x16x4_f32(
              false, a, false, b, (short)0, acc[t], false, false);
        }
      }
      float* out = Os[w];
      #pragma unroll
      for (int t = 0; t < 8; ++t)
        #pragma unroll
        for (int j = 0; j < 8; ++j)
          out[(long)(tile * 16 + rrow + j) * OUT_DIM + t * 16 + m] = acc[t][j];
    }
  }
}

// ---------------------------------------------------------------- edge fuse
// One wave per 16-edge tile: WMMA proj_e in registers, then per head-tile
// (== per 16x16 C tile) gather K[src]/Q[dst], form score, half-wave rowsum,
// exp(clip), write e_out, atomic scatter of V[src]*s and s (targets are
// L2-resident: 25.6MB << 192MB L2).
__global__ __launch_bounds__(128) void edge_kernel(
    const float* __restrict__ e,
    const int* __restrict__ src, const int* __restrict__ dst,
    const float* __restrict__ We,
    const float* __restrict__ Q, const float* __restrict__ K,
    const float* __restrict__ V,
    float* __restrict__ e_out, float* __restrict__ h_out,
    float* __restrict__ z)
{
  __shared__ float Wl[IN_DIM * OUT_DIM];   // 64 KB
  const int tid  = threadIdx.x;
  const int lane = tid & 31;
  const int wave = tid >> 5;
  const int m    = lane & 15;
  const int koff = (lane >> 4) << 1;
  const int half = lane >> 4;

  if (wave == 0) tdm_load_weight_to_lds(We, Wl);
  __syncthreads();

  const int tile = blockIdx.x * WAVES_PER_BLOCK + wave;
  const int e0   = tile * 16;
  if (e0 >= N_EDGES) return;

  // ---- proj_e = e_tile @ We  (8 C-tiles, one per head) ----
  v8f acc[8] = {};
  const long arow = (long)(e0 + m) * IN_DIM;
  #pragma unroll 4
  for (int k = 0; k < IN_DIM / 4; ++k) {
    const float2 a2 = *(const float2*)(e + arow + k * 4 + koff);
    v2f a; a[0] = a2.x; a[1] = a2.y;
    const int kb = (k * 4 + koff) * OUT_DIM;
    #pragma unroll
    for (int t = 0; t < 8; ++t) {
      const int col = t * 16 + m;
      v2f b; b[0] = Wl[kb + col]; b[1] = Wl[kb + OUT_DIM + col];
      acc[t] = __builtin_amdgcn_wmma_f32_16x16x4_f32(
          false, a, false, b, (short)0, acc[t], false, false);
    }
  }

  const float inv_sqrt_d = 0.25f;          // 1/sqrt(HEAD_DIM)
  #pragma unroll
  for (int t = 0; t < 8; ++t) {            // head / N-tile
    const int col = t * 16 + m;
    #pragma unroll
    for (int j = 0; j < 8; ++j) {          // C VGPR -> edge row
      const int row = j + half * 8;
      const int eg  = e0 + row;
      const int sn  = src[eg];
      const int dn  = dst[eg];
      const float kv = K[(long)sn * OUT_DIM + col];
      const float qv = Q[(long)dn * OUT_DIM + col];
      const float sc = kv * qv * inv_sqrt_d * acc[t][j];
      e_out[(long)eg * OUT_DIM + col] = sc;
      // sum the 16 columns of this head-row across the half-wave
      float sum = sc;
      #pragma unroll
      for (int o = 1; o < 16; o <<= 1) sum += __shfl_xor(sum, o, 32);
      sum = fminf(5.0f, fmaxf(-5.0f, sum));
      const float s = expf(sum);
      const float vv = V[(long)sn * OUT_DIM + col];
      atomicAdd(&h_out[(long)dn * OUT_DIM + col], vv * s);
      if (m == 0) atomicAdd(&z[(long)dn * HEADS + t], s);
    }
  }
}

// ---------------------------------------------------------------- normalize
__global__ void norm_kernel(float* __restrict__ h_out,
                            const float* __restrict__ z) {
  int i = blockIdx.x * blockDim.x + threadIdx.x;
  if (i < N_NODES * OUT_DIM) {
    const int n  = i >> 7;            // / 128
    const int hh = (i >> 4) & 7;      // head
    h_out[i] = h_out[i] / (z[n * HEADS + hh] + 1e-6f);
  }
}

// ---------------------------------------------------------------- launcher
extern "C" void kernel_launch(void* const* d_in, const int* in_sizes, int n_in,
                              void* d_out, int out_size, void* d_ws, size_t ws_size,
                              hipStream_t stream) {
  const float* h   = (const float*)d_in[0];
  const float* e   = (const float*)d_in[1];
  const int*   src = (const int*)d_in[2];
  const int*   dst = (const int*)d_in[3];
  const float* Wq  = (const float*)d_in[4];
  const float* Wk  = (const float*)d_in[5];
  const float* Wv  = (const float*)d_in[6];
  const float* We  = (const float*)d_in[7];

  float* out   = (float*)d_out;
  float* h_out = out;                                   // [N, 128]
  float* e_out = out + (long)N_NODES * OUT_DIM;         // [E, 128]

  float* Q = (float*)d_ws;
  float* K = Q + (long)N_NODES * OUT_DIM;
  float* V = K + (long)N_NODES * OUT_DIM;
  float* z = V + (long)N_NODES * OUT_DIM;               // [N, HEADS]

  // zero accumulation targets (d_out / d_ws are poisoned by the harness)
  {
    int n1 = N_NODES * OUT_DIM;
    zero_kernel<<<(n1 + 255) / 256, 256, 0, stream>>>(h_out, n1);
    int n2 = N_NODES * HEADS;
    zero_kernel<<<(n2 + 255) / 256, 256, 0, stream>>>(z, n2);
  }

  const int nodeTiles = (N_NODES + 15) / 16;            // 3125
  qkv_kernel<<<(nodeTiles + WAVES_PER_BLOCK - 1) / WAVES_PER_BLOCK, 128, 0, stream>>>(
      h, Wq, Wk, Wv, Q, K, V);

  const int edgeTiles = (N_EDGES + 15) / 16;            // 62500
  edge_kernel<<<(edgeTiles + WAVES_PER_BLOCK - 1) / WAVES_PER_BLOCK, 128, 0, stream>>>(
      e, src, dst, We, Q, K, V, e_out, h_out, z);

  norm_kernel<<<(N_NODES * OUT_DIM + 255) / 256, 256, 0, stream>>>(h_out, z);
}